// Loss_5334349381989
// MI455X (gfx1250) — compile-verified
//
#include <hip/hip_runtime.h>
#include <math.h>
#include <cstdint>
#include <utility>

#define AS1 __attribute__((address_space(1)))
#define AS3 __attribute__((address_space(3)))

typedef int v4i __attribute__((vector_size(16)));
typedef float f4 __attribute__((ext_vector_type(4)));

#ifdef __AMDGCN__
#if __has_builtin(__builtin_amdgcn_global_load_async_to_lds_b128)
#define HAVE_ASYNC_BUILTIN 1
#endif
#if __has_builtin(__builtin_amdgcn_s_wait_asynccnt)
#define HAVE_WAIT_BUILTIN 1
#endif
#endif

#if defined(HAVE_WAIT_BUILTIN)
#define WAITA(n) do { __builtin_amdgcn_s_wait_asynccnt(n); asm volatile("" ::: "memory"); } while (0)
#else
#define WAITA(n) asm volatile("s_wait_asynccnt " #n ::: "memory")
#endif

// one async 16B/lane copy, global -> LDS; OFF is a compile-time byte offset
// applied to BOTH addresses (ISA: INST_OFFSET is added on each side).
template <int OFF>
__device__ __forceinline__ void async_one(const float* g, float* l) {
#if defined(HAVE_ASYNC_BUILTIN)
  __builtin_amdgcn_global_load_async_to_lds_b128(
      (AS1 v4i*)(uintptr_t)g,
      (AS3 v4i*)(unsigned)(uintptr_t)l,
      OFF, /*cpol*/ 0);
#else
  unsigned ldso = (unsigned)(uintptr_t)l + (unsigned)OFF;
  const char* gp = (const char*)g + OFF;
  asm volatile("global_load_async_to_lds_b128 %0, %1, off"
               :: "v"(ldso), "v"(gp) : "memory");
#endif
}

// issue 25 async b128s covering one 400B row per lane
template <int... Js>
__device__ __forceinline__ void async_row(const float* g, float* l,
                                          std::integer_sequence<int, Js...>) {
  (async_one<16 * Js>(g, l), ...);
}

#define T_STEPS   1000
#define BATCH     256
#define NC        100
#define CHUNKS    4      // chunks per batch element -> grid = 256*4
#define ROWS_PB   250    // rows (t values) per block
#define BWAVES    2      // waves per block (64 threads)
#define STAGE_R   64     // rows per stage (BWAVES * 32)
#define L2E       1.4426950408889634f
#define LN2       0.6931471805599453f
#define LOG_T     6.907755278982137f   // ln(1000)

__global__ __launch_bounds__(64) void mc_logsumexp_kernel(
    const float* __restrict__ mu, const float* __restrict__ ls,
    const float* __restrict__ eps, const int* __restrict__ y,
    float* __restrict__ partM, float* __restrict__ partS)
{
  // double-buffered: 2 x 2 waves x 32 rows x 100 floats = 51.2 KB
  __shared__ float ebuf[2][BWAVES][32][NC];
  __shared__ __attribute__((aligned(16))) float pm2[NC];  // mu * log2(e)
  __shared__ __attribute__((aligned(16))) float ps2[NC];  // sigma * log2(e)
  __shared__ float redM[BWAVES], redS[BWAVES];

  const int b     = blockIdx.x >> 2;     // batch element
  const int chunk = blockIdx.x & 3;      // t-range chunk
  const int t0    = chunk * ROWS_PB;
  const int tid   = threadIdx.x;
  const int lane  = tid & 31;
  const int w     = tid >> 5;

  // ---- stage issue: lane i fetches row (s*64 + w*32 + i); 25 async b128s ----
  auto issue = [&](int s) {
    int rl = s * STAGE_R + (w << 5) + lane;
    int t  = t0 + (rl < ROWS_PB ? rl : ROWS_PB - 1);   // clamp tail (in-bounds dup)
    const float* g = eps + (size_t)(t * BATCH + b) * NC;
    float* l = &ebuf[s & 1][w][lane][0];
    async_row(g, l, std::make_integer_sequence<int, 25>{});
  };

  // kick off the first two stages before doing any setup work
  issue(0);
  issue(1);

  // ---- per-block class parameter tables (log2 domain) ----
  for (int c = tid; c < NC; c += 64) {
    pm2[c] = mu[b * NC + c] * L2E;
    ps2[c] = __builtin_amdgcn_exp2f(0.5f * L2E * ls[b * NC + c]) * L2E;
  }
  __syncthreads();

  const int   yb  = y[b];
  const float muY = pm2[yb];
  const float sgY = ps2[yb];

  float accM = -INFINITY, accS = 0.0f;   // per-thread online logsumexp (base 2)

  // ---- consume one stage: each lane processes its own 100-class row ----
  auto consume = [&](int s) {
    const float* row = &ebuf[s & 1][w][lane][0];
    f4 lg[25];
#pragma unroll
    for (int j = 0; j < 25; ++j) {
      f4 e  = *(const f4*)(row + 4 * j);
      f4 m  = *(const f4*)(&pm2[4 * j]);   // broadcast read (conflict-free)
      f4 sg = *(const f4*)(&ps2[4 * j]);
      lg[j] = m + sg * e;                  // fma-contracted, log2-domain logits
    }
    f4 mx = lg[0];
#pragma unroll
    for (int j = 1; j < 25; ++j) {
      mx.x = fmaxf(mx.x, lg[j].x); mx.y = fmaxf(mx.y, lg[j].y);
      mx.z = fmaxf(mx.z, lg[j].z); mx.w = fmaxf(mx.w, lg[j].w);
    }
    float mr = fmaxf(fmaxf(mx.x, mx.y), fmaxf(mx.z, mx.w));
    f4 acc = {0.f, 0.f, 0.f, 0.f};
#pragma unroll
    for (int j = 0; j < 25; ++j) {
      acc.x += __builtin_amdgcn_exp2f(lg[j].x - mr);
      acc.y += __builtin_amdgcn_exp2f(lg[j].y - mr);
      acc.z += __builtin_amdgcn_exp2f(lg[j].z - mr);
      acc.w += __builtin_amdgcn_exp2f(lg[j].w - mr);
    }
    float ssum = (acc.x + acc.y) + (acc.z + acc.w);
    float lse  = mr + __builtin_amdgcn_logf(ssum);        // v_log_f32 = log2
    float ly   = fmaf(sgY, row[yb], muY);                 // true-class logit
    float r    = ly - lse;                                // log2 p_t(y)
    int   rl   = s * STAGE_R + (w << 5) + lane;
    if (rl < ROWS_PB) {                                   // mask clamped tail rows
      float nm = fmaxf(accM, r);
      accS = accS * __builtin_amdgcn_exp2f(accM - nm) + __builtin_amdgcn_exp2f(r - nm);
      accM = nm;
    }
  };

  // ---- 4-stage pipeline; each stage = 25 async loads in flight ----
  WAITA(25); consume(0); issue(2);
  WAITA(25); consume(1); issue(3);
  WAITA(25); consume(2);
  WAITA(0);  consume(3);

  // ---- merge (M,S) across the 32 lanes of each wave ----
#pragma unroll
  for (int off = 16; off >= 1; off >>= 1) {
    float Mo = __shfl_xor(accM, off, 32);
    float So = __shfl_xor(accS, off, 32);
    float nm = fmaxf(accM, Mo);
    accS = accS * __builtin_amdgcn_exp2f(accM - nm) + So * __builtin_amdgcn_exp2f(Mo - nm);
    accM = nm;
  }
  if (lane == 0) { redM[w] = accM; redS[w] = accS; }
  __syncthreads();
  if (tid == 0) {
    float M = fmaxf(redM[0], redM[1]);
    float S = redS[0] * __builtin_amdgcn_exp2f(redM[0] - M)
            + redS[1] * __builtin_amdgcn_exp2f(redM[1] - M);
    partM[blockIdx.x] = M;
    partS[blockIdx.x] = S;
  }
}

__global__ __launch_bounds__(256) void final_reduce_kernel(
    const float* __restrict__ partM, const float* __restrict__ partS,
    float* __restrict__ out)
{
  __shared__ float sm[8];
  const int b    = threadIdx.x;          // one thread per batch element
  const int lane = b & 31;
  const int w    = b >> 5;

  float M = -INFINITY, S = 0.0f;
#pragma unroll
  for (int j = 0; j < CHUNKS; ++j) {
    float Mj = partM[b * CHUNKS + j];
    float Sj = partS[b * CHUNKS + j];
    float nm = fmaxf(M, Mj);
    S = S * __builtin_amdgcn_exp2f(M - nm) + Sj * __builtin_amdgcn_exp2f(Mj - nm);
    M = nm;
  }
  float picked = (M + __builtin_amdgcn_logf(S)) * LN2 - LOG_T;  // ln p(y|b)

  float v = picked;
#pragma unroll
  for (int off = 16; off >= 1; off >>= 1) v += __shfl_xor(v, off, 32);
  if (lane == 0) sm[w] = v;
  __syncthreads();
  if (b == 0) {
    float s = 0.0f;
#pragma unroll
    for (int i = 0; i < 8; ++i) s += sm[i];
    out[0] = -s * (1.0f / 256.0f);
  }
}

extern "C" void kernel_launch(void* const* d_in, const int* in_sizes, int n_in,
                              void* d_out, int out_size, void* d_ws, size_t ws_size,
                              hipStream_t stream) {
  (void)in_sizes; (void)n_in; (void)out_size; (void)ws_size;
  const float* mu  = (const float*)d_in[0];
  const float* ls  = (const float*)d_in[1];
  const float* eps = (const float*)d_in[2];
  const int*   y   = (const int*)d_in[3];
  float* partM = (float*)d_ws;                  // 1024 floats
  float* partS = (float*)d_ws + BATCH * CHUNKS; // next 1024 floats
  float* out   = (float*)d_out;

  mc_logsumexp_kernel<<<BATCH * CHUNKS, 64, 0, stream>>>(mu, ls, eps, y, partM, partS);
  final_reduce_kernel<<<1, 256, 0, stream>>>(partM, partS, out);
}